// SpectralBlock_5274219840159
// MI455X (gfx1250) — compile-verified
//
#include <hip/hip_runtime.h>
#include <math.h>

typedef float v2f __attribute__((ext_vector_type(2)));
typedef float v8f __attribute__((ext_vector_type(8)));

#define NB 8
#define NH 256
#define NW 256
#define NC 64
#define TWO_PI 6.283185307179586f

// D = A(16x4) * B(4x16) + C, f32 WMMA (CDNA5)
__device__ __forceinline__ v8f wmma4(v8f acc, v2f a, v2f b) {
  return __builtin_amdgcn_wmma_f32_16x16x4_f32(false, a, false, b, (short)0, acc,
                                               false, false);
}

// ---------------------------------------------------------------------------
// Stage 1 (fused): per (b,h) row, truncated DFT along W (ky=0..15) via WMMA,
// and the dense channel-mix skip y_p = x @ K + bias (written to d_out).
// ---------------------------------------------------------------------------
__global__ __launch_bounds__(256) void k_dftw_dense(
    const float* __restrict__ x, const float* __restrict__ kmatg,
    const float* __restrict__ biasg, float* __restrict__ yout,
    float* __restrict__ x1r, float* __restrict__ x1i) {
  __shared__ float xch[64 * 64];   // x chunk [w(64)][c(64)]
  __shared__ float t1r[16 * 64];   // twiddle cos  [ky][w]
  __shared__ float t1i[16 * 64];   // twiddle -sin [ky][w]
  __shared__ float kmat[64 * 64];  // dense_k [cin][cout]
  __shared__ float bias[64];

  const int tid = threadIdx.x;
  const int wave = tid >> 5;
  const int lane = tid & 31;
  const int half = lane >> 4;
  const int l16 = lane & 15;
  const int b = blockIdx.x >> 8;
  const int h = blockIdx.x & 255;

  for (int i = tid; i < 64 * 64; i += 256) kmat[i] = kmatg[i];
  if (tid < 64) bias[tid] = biasg[tid];

  const int part = wave >> 2;  // 0 = real, 1 = imag DFT tile
  const int nt = wave & 3;     // DFT channel tile
  v8f accdft = {};

  const float* xrow = x + ((size_t)(b * NH + h)) * NW * NC;
  float* yrow = yout + ((size_t)(b * NH + h)) * NW * NC;

  for (int ch = 0; ch < 4; ++ch) {
    __syncthreads();
    for (int i = tid; i < 64 * 64; i += 256)
      xch[i] = xrow[(size_t)ch * 64 * 64 + i];
    for (int i = tid; i < 16 * 64; i += 256) {
      int ky = i >> 6, wl = i & 63;
      int m = ((ch * 64 + wl) * ky) & 255;
      float s, c;
      __sincosf(-TWO_PI * (float)m * (1.0f / 256.0f), &s, &c);
      t1r[i] = c;
      t1i[i] = s;
    }
    __syncthreads();

    // DFT-W partial: A = twiddle[16ky x 64w], B = xch[64w x 16c]
    const float* t1p = part ? t1i : t1r;
#pragma unroll
    for (int k0 = 0; k0 < 64; k0 += 4) {
      int k = k0 + 2 * half;
      v2f a = {t1p[l16 * 64 + k], t1p[l16 * 64 + k + 1]};
      v2f bb = {xch[k * 64 + nt * 16 + l16], xch[(k + 1) * 64 + nt * 16 + l16]};
      accdft = wmma4(accdft, a, bb);
    }

    // Dense skip for this w-chunk: 16 tiles, 2 per wave
#pragma unroll
    for (int t = 0; t < 2; ++t) {
      int tile = wave + t * 8;
      int mt = tile >> 2, nt2 = tile & 3;
      v8f acc = {};
#pragma unroll
      for (int k0 = 0; k0 < 64; k0 += 4) {
        int k = k0 + 2 * half;
        v2f a = {xch[(mt * 16 + l16) * 64 + k], xch[(mt * 16 + l16) * 64 + k + 1]};
        v2f bb = {kmat[k * 64 + nt2 * 16 + l16], kmat[(k + 1) * 64 + nt2 * 16 + l16]};
        acc = wmma4(acc, a, bb);
      }
      int col = nt2 * 16 + l16;
      float bv = bias[col];
#pragma unroll
      for (int v = 0; v < 8; ++v) {
        int wrow = ch * 64 + mt * 16 + v + 8 * half;
        yrow[wrow * NC + col] = acc[v] + bv;
      }
    }
  }

  // Store X1[b][h][ky][c] re/im
  float* xp = part ? x1i : x1r;
#pragma unroll
  for (int v = 0; v < 8; ++v) {
    int ky = v + 8 * half;
    xp[(((size_t)(b * NH + h)) * 16 + ky) * 64 + nt * 16 + l16] = accdft[v];
  }
}

// ---------------------------------------------------------------------------
// Stage 2: DFT along H for kx in {0..15, 240..255}; complex GEMM via WMMA.
// One block per (b, ky).
// ---------------------------------------------------------------------------
__global__ __launch_bounds__(256) void k_dfth(
    const float* __restrict__ x1r, const float* __restrict__ x1i,
    float* __restrict__ x2r, float* __restrict__ x2i) {
  __shared__ float lr[64 * 64];   // X1r chunk [h(64)][c]
  __shared__ float li[64 * 64];   // X1i chunk
  __shared__ float t2r[32 * 64];  // cos  [j][h]
  __shared__ float t2i[32 * 64];  // -sin [j][h]
  const int tid = threadIdx.x;
  const int wave = tid >> 5;
  const int lane = tid & 31;
  const int half = lane >> 4;
  const int l16 = lane & 15;
  const int b = blockIdx.x >> 4;
  const int ky = blockIdx.x & 15;

  v8f acc[2] = {{}, {}};  // 2 tiles per wave

  for (int ch = 0; ch < 4; ++ch) {
    __syncthreads();
    for (int i = tid; i < 64 * 64; i += 256) {
      int hh = i >> 6, c = i & 63;
      size_t src = (((size_t)b * NH + ch * 64 + hh) * 16 + ky) * 64 + c;
      lr[i] = x1r[src];
      li[i] = x1i[src];
    }
    for (int i = tid; i < 32 * 64; i += 256) {
      int j = i >> 6, hh = i & 63;
      int kx = (j < 16) ? j : (j + 224);
      int m = ((ch * 64 + hh) * kx) & 255;
      float s, c;
      __sincosf(-TWO_PI * (float)m * (1.0f / 256.0f), &s, &c);
      t2r[i] = c;
      t2i[i] = s;
    }
    __syncthreads();
#pragma unroll
    for (int t = 0; t < 2; ++t) {
      int tile = wave * 2 + t;
      int p = tile >> 3;          // 0=real out, 1=imag out
      int mt = (tile >> 2) & 1;   // kx tile
      int nt = tile & 3;          // channel tile
      const float* bmain = p ? li : lr;  // T2r @ (p ? X1i : X1r)
      const float* bsec = p ? lr : li;   // +/- T2i @ (p ? X1r : X1i)
      float sgn = p ? 1.0f : -1.0f;
#pragma unroll
      for (int k0 = 0; k0 < 64; k0 += 4) {
        int k = k0 + 2 * half;
        int arow = (mt * 16 + l16) * 64;
        v2f a1 = {t2r[arow + k], t2r[arow + k + 1]};
        v2f b1 = {bmain[k * 64 + nt * 16 + l16], bmain[(k + 1) * 64 + nt * 16 + l16]};
        acc[t] = wmma4(acc[t], a1, b1);
        v2f a2 = {sgn * t2i[arow + k], sgn * t2i[arow + k + 1]};
        v2f b2 = {bsec[k * 64 + nt * 16 + l16], bsec[(k + 1) * 64 + nt * 16 + l16]};
        acc[t] = wmma4(acc[t], a2, b2);
      }
    }
  }
#pragma unroll
  for (int t = 0; t < 2; ++t) {
    int tile = wave * 2 + t;
    int p = tile >> 3;
    int mt = (tile >> 2) & 1;
    int nt = tile & 3;
    float* outp = p ? x2i : x2r;
#pragma unroll
    for (int v = 0; v < 8; ++v) {
      int j = mt * 16 + v + 8 * half;
      outp[(((size_t)b * 32 + j) * 16 + ky) * 64 + nt * 16 + l16] = acc[t][v];
    }
  }
}

// ---------------------------------------------------------------------------
// Stage 3: per-mode complex channel mixing, Y2 = X2 @ (Wr + i Wi). M=8 only,
// so VALU (tiny FLOP count). One block per mode (j, ky), 512 threads = (b,c).
// ---------------------------------------------------------------------------
__global__ __launch_bounds__(512) void k_modes(
    const float* __restrict__ x2r, const float* __restrict__ x2i,
    const float* __restrict__ w1r, const float* __restrict__ w1i,
    const float* __restrict__ w2r, const float* __restrict__ w2i,
    float* __restrict__ y2r, float* __restrict__ y2i) {
  __shared__ float xr[8 * 64];
  __shared__ float xi[8 * 64];
  const int tid = threadIdx.x;
  const int j = blockIdx.x >> 4;
  const int ky = blockIdx.x & 15;
  const int tb = tid >> 6;
  const int tc = tid & 63;
  {
    size_t src = (((size_t)tb * 32 + j) * 16 + ky) * 64 + tc;
    xr[tid] = x2r[src];
    xi[tid] = x2i[src];
  }
  __syncthreads();
  const float* wr = (j < 16) ? w1r : w2r;
  const float* wi = (j < 16) ? w1i : w2i;
  const int mxw = (j < 16) ? j : (j - 16);
  const float* wrb = wr + (((size_t)mxw * 16 + ky) * 64) * 64 + tc;
  const float* wib = wi + (((size_t)mxw * 16 + ky) * 64) * 64 + tc;
  float ar = 0.f, ai = 0.f;
#pragma unroll 8
  for (int ci = 0; ci < 64; ++ci) {
    float wrv = wrb[ci * 64];
    float wiv = wib[ci * 64];
    float xrv = xr[tb * 64 + ci];
    float xiv = xi[tb * 64 + ci];
    ar += xrv * wrv - xiv * wiv;
    ai += xrv * wiv + xiv * wrv;
  }
  size_t dst = (((size_t)tb * 32 + j) * 16 + ky) * 64 + tc;
  y2r[dst] = ar;
  y2i[dst] = ai;
}

// ---------------------------------------------------------------------------
// Stage 4: inverse DFT along H (sum over 32 kx modes), 1/(H*W) folded here.
// One block per (b, ky). Twiddles generated per lane (TRANS co-exec w/ WMMA).
// ---------------------------------------------------------------------------
__global__ __launch_bounds__(256) void k_idfth(
    const float* __restrict__ y2r, const float* __restrict__ y2i,
    float* __restrict__ y1r, float* __restrict__ y1i) {
  __shared__ float lr[32 * 64];
  __shared__ float li[32 * 64];
  const int tid = threadIdx.x;
  const int wave = tid >> 5;
  const int lane = tid & 31;
  const int half = lane >> 4;
  const int l16 = lane & 15;
  const int b = blockIdx.x >> 4;
  const int ky = blockIdx.x & 15;

  for (int i = tid; i < 32 * 64; i += 256) {
    int j = i >> 6, c = i & 63;
    size_t src = (((size_t)b * 32 + j) * 16 + ky) * 64 + c;
    lr[i] = y2r[src];
    li[i] = y2i[src];
  }
  __syncthreads();
  const float scale = 1.0f / 65536.0f;  // 1/(H*W)
  for (int mt = wave; mt < 16; mt += 8) {
    const int hrowA = mt * 16 + l16;
    float er[16], ei[16];
#pragma unroll
    for (int s = 0; s < 8; ++s) {
#pragma unroll
      for (int d = 0; d < 2; ++d) {
        int j = s * 4 + 2 * half + d;
        int kx = (j < 16) ? j : (j + 224);
        int m = (hrowA * kx) & 255;
        float sn, cs;
        __sincosf(TWO_PI * (float)m * (1.0f / 256.0f), &sn, &cs);
        er[s * 2 + d] = cs;
        ei[s * 2 + d] = sn;
      }
    }
#pragma unroll
    for (int p = 0; p < 2; ++p) {
      const float* bmain = p ? li : lr;
      const float* bsec = p ? lr : li;
      float sgn = p ? 1.0f : -1.0f;
#pragma unroll
      for (int nt = 0; nt < 4; ++nt) {
        v8f acc = {};
#pragma unroll
        for (int s = 0; s < 8; ++s) {
          int k = s * 4 + 2 * half;
          v2f a1 = {er[s * 2], er[s * 2 + 1]};
          v2f b1 = {bmain[k * 64 + nt * 16 + l16], bmain[(k + 1) * 64 + nt * 16 + l16]};
          acc = wmma4(acc, a1, b1);
          v2f a2 = {sgn * ei[s * 2], sgn * ei[s * 2 + 1]};
          v2f b2 = {bsec[k * 64 + nt * 16 + l16], bsec[(k + 1) * 64 + nt * 16 + l16]};
          acc = wmma4(acc, a2, b2);
        }
        float* outp = p ? y1i : y1r;
#pragma unroll
        for (int v = 0; v < 8; ++v) {
          int hrow = mt * 16 + v + 8 * half;
          outp[(((size_t)b * NH + hrow) * 16 + ky) * 64 + nt * 16 + l16] =
              acc[v] * scale;
        }
      }
    }
  }
}

// ---------------------------------------------------------------------------
// Stage 5: inverse real DFT along W (Hermitian, ky weights {1,2}), add skip
// y_p (in d_out), write y, emit deterministic per-(b,h) channel partial sums.
// ---------------------------------------------------------------------------
__global__ __launch_bounds__(256) void k_idftw_finish(
    const float* __restrict__ y1r, const float* __restrict__ y1i,
    float* __restrict__ y, float* __restrict__ pp1, float* __restrict__ pp2) {
  __shared__ float lr[16 * 64];
  __shared__ float li[16 * 64];
  __shared__ float ldsP[16 * 64];   // per (wave,half) buckets
  __shared__ float ldsP2[16 * 64];
  const int tid = threadIdx.x;
  const int wave = tid >> 5;
  const int lane = tid & 31;
  const int half = lane >> 4;
  const int l16 = lane & 15;
  const int b = blockIdx.x >> 8;
  const int h = blockIdx.x & 255;

  for (int i = tid; i < 16 * 64; i += 256) {
    size_t src = (((size_t)b * NH + h) * 16) * 64 + i;
    lr[i] = y1r[src];
    li[i] = y1i[src];
  }
  __syncthreads();

  float* yrow = y + ((size_t)(b * NH + h)) * NW * NC;
  float sv[4] = {0.f, 0.f, 0.f, 0.f}, sv2[4] = {0.f, 0.f, 0.f, 0.f};

#pragma unroll
  for (int mt2 = 0; mt2 < 2; ++mt2) {
    int mt = wave * 2 + mt2;
    int wrowA = mt * 16 + l16;
    float a1v[8], a2v[8];
#pragma unroll
    for (int s = 0; s < 4; ++s) {
#pragma unroll
      for (int d = 0; d < 2; ++d) {
        int ky = s * 4 + 2 * half + d;
        float wk = (ky == 0) ? 1.0f : 2.0f;
        int m = (wrowA * ky) & 255;
        float sn, cs;
        __sincosf(TWO_PI * (float)m * (1.0f / 256.0f), &sn, &cs);
        a1v[s * 2 + d] = wk * cs;
        a2v[s * 2 + d] = -wk * sn;
      }
    }
#pragma unroll
    for (int nt = 0; nt < 4; ++nt) {
      v8f acc = {};
#pragma unroll
      for (int s = 0; s < 4; ++s) {
        int k = s * 4 + 2 * half;
        v2f aa1 = {a1v[s * 2], a1v[s * 2 + 1]};
        v2f bb1 = {lr[k * 64 + nt * 16 + l16], lr[(k + 1) * 64 + nt * 16 + l16]};
        acc = wmma4(acc, aa1, bb1);
        v2f aa2 = {a2v[s * 2], a2v[s * 2 + 1]};
        v2f bb2 = {li[k * 64 + nt * 16 + l16], li[(k + 1) * 64 + nt * 16 + l16]};
        acc = wmma4(acc, aa2, bb2);
      }
      int col = nt * 16 + l16;
#pragma unroll
      for (int v = 0; v < 8; ++v) {
        int wrow = mt * 16 + v + 8 * half;
        float val = acc[v] + yrow[wrow * NC + col];
        yrow[wrow * NC + col] = val;
        sv[nt] += val;
        sv2[nt] += val * val;
      }
    }
  }
#pragma unroll
  for (int nt = 0; nt < 4; ++nt) {
    ldsP[(wave * 2 + half) * 64 + nt * 16 + l16] = sv[nt];
    ldsP2[(wave * 2 + half) * 64 + nt * 16 + l16] = sv2[nt];
  }
  __syncthreads();
  if (tid < 64) {
    float t1 = 0.f, t2 = 0.f;
    for (int q = 0; q < 16; ++q) {
      t1 += ldsP[q * 64 + tid];
      t2 += ldsP2[q * 64 + tid];
    }
    pp1[((size_t)b * NH + h) * 64 + tid] = t1;
    pp2[((size_t)b * NH + h) * 64 + tid] = t2;
  }
}

// ---------------------------------------------------------------------------
// Stage 6: reduce per-(b,h) partials over h -> per-(b,c) sum / sumsq.
// ---------------------------------------------------------------------------
__global__ __launch_bounds__(256) void k_reduce(
    const float* __restrict__ pp1, const float* __restrict__ pp2,
    float* __restrict__ gsum, float* __restrict__ gsumsq) {
  __shared__ float r1[256];
  __shared__ float r2[256];
  const int b = blockIdx.x >> 6;
  const int c = blockIdx.x & 63;
  const int t = threadIdx.x;
  r1[t] = pp1[((size_t)b * NH + t) * 64 + c];
  r2[t] = pp2[((size_t)b * NH + t) * 64 + c];
  __syncthreads();
  for (int s = 128; s > 0; s >>= 1) {
    if (t < s) {
      r1[t] += r1[t + s];
      r2[t] += r2[t + s];
    }
    __syncthreads();
  }
  if (t == 0) {
    gsum[b * 64 + c] = r1[0];
    gsumsq[b * 64 + c] = r2[0];
  }
}

// ---------------------------------------------------------------------------
// Stage 7: instance norm + tanh-GELU, in place on d_out.
// ---------------------------------------------------------------------------
__global__ __launch_bounds__(256) void k_norm_gelu(
    float* __restrict__ y, const float* __restrict__ gsum,
    const float* __restrict__ gsumsq, const float* __restrict__ nsc,
    const float* __restrict__ nb) {
  const size_t i = (size_t)blockIdx.x * 256 + threadIdx.x;
  const int c = (int)(i & 63);
  const int b = (int)(i >> 22);  // H*W*C = 2^22
  const float inv_n = 1.0f / 65536.0f;
  float mean = gsum[b * 64 + c] * inv_n;
  float var = gsumsq[b * 64 + c] * inv_n - mean * mean;
  float inv = rsqrtf(var + 1e-6f);
  float v = (y[i] - mean) * inv * nsc[c] + nb[c];
  float t = tanhf(0.7978845608028654f * (v + 0.044715f * v * v * v));
  y[i] = 0.5f * v * (1.0f + t);
}

extern "C" void kernel_launch(void* const* d_in, const int* in_sizes, int n_in,
                              void* d_out, int out_size, void* d_ws,
                              size_t ws_size, hipStream_t stream) {
  (void)in_sizes; (void)n_in; (void)out_size; (void)ws_size;
  const float* x = (const float*)d_in[0];
  const float* w1r = (const float*)d_in[1];
  const float* w1i = (const float*)d_in[2];
  const float* w2r = (const float*)d_in[3];
  const float* w2i = (const float*)d_in[4];
  const float* dk = (const float*)d_in[5];
  const float* db = (const float*)d_in[6];
  const float* nsc = (const float*)d_in[7];
  const float* nb = (const float*)d_in[8];
  float* out = (float*)d_out;
  float* ws = (float*)d_ws;

  // ws layout (floats)
  float* gsum = ws;                       // 512
  float* gsumsq = gsum + 512;             // 512
  float* x1r = gsumsq + 512;              // B*H*16*64 = 2097152
  float* x1i = x1r + 2097152;             // 2097152
  float* x2r = x1i + 2097152;             // B*32*16*64 = 262144
  float* x2i = x2r + 262144;
  float* y2r = x2i + 262144;
  float* y2i = y2r + 262144;
  float* pp1 = y2i + 262144;              // B*H*64 = 131072
  float* pp2 = pp1 + 131072;              // total ~22 MB
  float* y1r = x1r;  // X1 dead after stage 2 -> alias
  float* y1i = x1i;

  k_dftw_dense<<<NB * NH, 256, 0, stream>>>(x, dk, db, out, x1r, x1i);
  k_dfth<<<NB * 16, 256, 0, stream>>>(x1r, x1i, x2r, x2i);
  k_modes<<<32 * 16, 512, 0, stream>>>(x2r, x2i, w1r, w1i, w2r, w2i, y2r, y2i);
  k_idfth<<<NB * 16, 256, 0, stream>>>(y2r, y2i, y1r, y1i);
  k_idftw_finish<<<NB * NH, 256, 0, stream>>>(y1r, y1i, out, pp1, pp2);
  k_reduce<<<NB * 64, 256, 0, stream>>>(pp1, pp2, gsum, gsumsq);
  const unsigned total_blocks = (unsigned)(((size_t)NB * NH * NW * NC) / 256);
  k_norm_gelu<<<total_blocks, 256, 0, stream>>>(out, gsum, gsumsq, nsc, nb);
}